// GADBase_50345606644199
// MI455X (gfx1250) — compile-verified
//
#include <hip/hip_runtime.h>

// GADBase guided-depth-diffusion for MI455X (gfx1250, wave32).
//
// Strategy:
//  * compute_c's rfft2/high-pass/irfft2 is linear and channel-mean'd -> run it
//    ONCE on the channel-averaged plane, implemented as 12 GEMMs (512^3) with
//    native f32 WMMA (v_wmma_f32_16x16x4_f32).
//  * diffuse_step's frequency branch is a fp32 no-op (sigma=0.1 gaussian on
//    |F| is a delta kernel: weights [2e-22, 1, 2e-22]); only the anisotropic
//    stencil + adjust_step pooling remain in the 64-iteration loop.

#define W512 512
#define NPIX (512 * 512)
#define LSTEP 0.24f
#define KKINV (1.0f / (0.03f * 0.03f))
#define TWO_PI 6.28318530717958647692f

typedef float v2f __attribute__((ext_vector_type(2)));
typedef float v8f __attribute__((ext_vector_type(8)));

// ---------------------------------------------------------------- WMMA GEMM
// Out = alpha * (s1*A1*B1 [+ s2*A2*B2]); all matrices 512x512 row-major f32.
// One 16x16 output tile per wave; grid (32,8) x 128 threads = 1024 tiles.
template <int USE2>
__global__ __launch_bounds__(128) void gemm2_wmma(
    const float* __restrict__ A1, const float* __restrict__ B1, float s1,
    const float* __restrict__ A2, const float* __restrict__ B2, float s2,
    float* __restrict__ Out, float alpha) {
  const int lane = threadIdx.x & 31;
  const int wave = threadIdx.x >> 5;
  const int tileM = ((blockIdx.y << 2) + wave) << 4;
  const int tileN = blockIdx.x << 4;
  const int l15 = lane & 15;
  const int half = lane >> 4;   // 0: lanes 0-15, 1: lanes 16-31
  const int koff = half << 1;   // ISA A-frag: VGPR0/1 = K0,K1 (lo lanes) K2,K3 (hi lanes)
  const int mrow = tileM + l15;
  const int ncol = tileN + l15;

  v8f acc1, acc2;
#pragma unroll
  for (int r = 0; r < 8; ++r) { acc1[r] = 0.0f; acc2[r] = 0.0f; }

  const float* a1p = A1 + mrow * W512 + koff;
  const float* b1p = B1 + koff * W512 + ncol;
  const float* a2p = USE2 ? (A2 + mrow * W512 + koff) : nullptr;
  const float* b2p = USE2 ? (B2 + koff * W512 + ncol) : nullptr;

  for (int k = 0; k < W512; k += 4) {
    v2f a1 = *(const v2f*)(a1p + k);                  // A[m][k+koff .. +1]
    v2f b1;
    b1.x = b1p[k * W512];                             // B[k+koff][n]
    b1.y = b1p[(k + 1) * W512];                       // B[k+koff+1][n]
    acc1 = __builtin_amdgcn_wmma_f32_16x16x4_f32(false, a1, false, b1,
                                                 (short)0, acc1, false, false);
    if (USE2) {
      v2f a2 = *(const v2f*)(a2p + k);
      v2f b2;
      b2.x = b2p[k * W512];
      b2.y = b2p[(k + 1) * W512];
      acc2 = __builtin_amdgcn_wmma_f32_16x16x4_f32(false, a2, false, b2,
                                                   (short)0, acc2, false, false);
    }
  }
#pragma unroll
  for (int r = 0; r < 8; ++r) {                        // C/D: VGPR r -> rows r, r+8
    int row = tileM + r + (half << 3);
    float v = s1 * acc1[r];
    if (USE2) v += s2 * acc2[r];
    Out[row * W512 + ncol] = alpha * v;
  }
}

// ---------------------------------------------------------------- small kernels
__global__ void kmin_shift(const float* __restrict__ src, float* __restrict__ shift) {
  __shared__ float s[256];
  float m = 3.4e38f;
  for (int i = threadIdx.x; i < 4096; i += 256) m = fminf(m, src[i]);
  s[threadIdx.x] = m;
  __syncthreads();
  for (int off = 128; off > 0; off >>= 1) {
    if ((int)threadIdx.x < off) s[threadIdx.x] = fminf(s[threadIdx.x], s[threadIdx.x + off]);
    __syncthreads();
  }
  if (threadIdx.x == 0) shift[0] = (s[0] <= 0.1f) ? 0.1f : 0.0f;
}

__global__ void kprep(const float* __restrict__ g, const float* __restrict__ yb,
                      const float* __restrict__ shift, float* __restrict__ imgA,
                      float* __restrict__ ibar) {
  int i = blockIdx.x * blockDim.x + threadIdx.x;
  float sh = shift[0];
  float im = yb[i] + sh;
  imgA[i] = im;
  ibar[i] = 0.25f * (g[i] + g[i + NPIX] + g[i + 2 * NPIX] + im);
}

__global__ void ksrc(const float* __restrict__ src, const float* __restrict__ msk,
                     const float* __restrict__ shift, float* __restrict__ srcs,
                     float* __restrict__ keep) {
  int i = blockIdx.x * blockDim.x + threadIdx.x;
  srcs[i] = src[i] + shift[0];
  keep[i] = (msk[i] < 0.5f) ? 0.0f : 1.0f;
}

__global__ void kdiffs(const float* __restrict__ g, const float* __restrict__ imgA,
                       float* __restrict__ Av, float* __restrict__ Ah) {
  int i = blockIdx.x * blockDim.x + threadIdx.x;
  int y = i >> 9, x = i & 511;
  if (y < 511) {
    Av[i] = 0.25f * (fabsf(g[i + 512] - g[i]) +
                     fabsf(g[i + NPIX + 512] - g[i + NPIX]) +
                     fabsf(g[i + 2 * NPIX + 512] - g[i + 2 * NPIX]) +
                     fabsf(imgA[i + 512] - imgA[i]));
  }
  if (x < 511) {
    Ah[i] = 0.25f * (fabsf(g[i + 1] - g[i]) +
                     fabsf(g[i + NPIX + 1] - g[i + NPIX]) +
                     fabsf(g[i + 2 * NPIX + 1] - g[i + 2 * NPIX]) +
                     fabsf(imgA[i + 1] - imgA[i]));
  }
}

__global__ void ktwid(float* __restrict__ CM, float* __restrict__ SM) {
  int i = blockIdx.x * blockDim.x + threadIdx.x;
  int y = i >> 9, x = i & 511;
  int p = (y * x) & 511;  // exact phase reduction: avoids large-arg sincos error
  float ang = (float)p * (TWO_PI / 512.0f);
  CM[i] = cosf(ang);
  SM[i] = sinf(ang);
}

__global__ void kmask(float* __restrict__ Fr, float* __restrict__ Fi) {
  int i = blockIdx.x * blockDim.x + threadIdx.x;
  int u = i >> 9, v = i & 511;
  int dy, dx;
  if (v <= 256) { dy = u - 256; dx = v - 128; }
  else {        // Hermitian extension of the rfft-grid mask
    int u2 = (512 - u) & 511;
    int v2 = 512 - v;
    dy = u2 - 256; dx = v2 - 128;
  }
  if (dy * dy + dx * dx <= 128 * 128) { Fr[i] = 0.0f; Fi[i] = 0.0f; }
}

__global__ void kcoef(const float* __restrict__ Av, const float* __restrict__ Ah,
                      const float* __restrict__ E, float* __restrict__ cv,
                      float* __restrict__ ch) {
  int i = blockIdx.x * blockDim.x + threadIdx.x;
  int y = i >> 9, x = i & 511;
  if (y < 511) {
    float mv = Av[i] + E[i + 512];
    cv[i] = 1.0f / (1.0f + mv * mv * KKINV);
  }
  if (x < 511) {
    float mh = Ah[i] + E[i + 1];
    ch[i] = 1.0f / (1.0f + mh * mh * KKINV);
  }
}

__global__ void kstencil(const float* __restrict__ I, const float* __restrict__ cv,
                         const float* __restrict__ ch, float* __restrict__ Out) {
  int i = blockIdx.x * blockDim.x + threadIdx.x;
  int y = i >> 9, x = i & 511;
  float c = I[i];
  float out = c;  // alpha*I + (1-alpha)*I_freq == I (fp32 delta blur)
  if (y < 511) out += LSTEP * cv[i] * (I[i + 512] - c);
  if (y > 0)   out -= LSTEP * cv[i - 512] * (c - I[i - 512]);
  if (x < 511) out += LSTEP * ch[i] * (I[i + 1] - c);
  if (x > 0)   out -= LSTEP * ch[i - 1] * (c - I[i - 1]);
  Out[i] = out;
}

__global__ void kpool(const float* __restrict__ I, const float* __restrict__ srcs,
                      const float* __restrict__ keep, float* __restrict__ ratio) {
  __shared__ float s[64];
  int bx = blockIdx.x, by = blockIdx.y;
  int ly = threadIdx.x >> 3, lx = threadIdx.x & 7;
  s[threadIdx.x] = I[(by * 8 + ly) * 512 + bx * 8 + lx];
  __syncthreads();
  for (int off = 32; off > 0; off >>= 1) {
    if ((int)threadIdx.x < off) s[threadIdx.x] += s[threadIdx.x + off];
    __syncthreads();
  }
  if (threadIdx.x == 0) {
    float mean = s[0] * (1.0f / 64.0f);
    int j = by * 64 + bx;
    ratio[j] = (keep[j] != 0.0f) ? srcs[j] / (mean + 1e-8f) : 1.0f;
  }
}

__global__ void kscale(const float* __restrict__ I, const float* __restrict__ ratio,
                       float* __restrict__ Out) {
  int i = blockIdx.x * blockDim.x + threadIdx.x;
  int y = i >> 9, x = i & 511;
  Out[i] = I[i] * ratio[(y >> 3) * 64 + (x >> 3)];
}

__global__ void kfinal(const float* __restrict__ I, const float* __restrict__ shift,
                       float* __restrict__ out) {
  int i = blockIdx.x * blockDim.x + threadIdx.x;
  out[i] = I[i] - shift[0];
}

// ---------------------------------------------------------------- launcher
extern "C" void kernel_launch(void* const* d_in, const int* in_sizes, int n_in,
                              void* d_out, int out_size, void* d_ws, size_t ws_size,
                              hipStream_t stream) {
  const float* guide = (const float*)d_in[0];   // [3,512,512]
  const float* ybic  = (const float*)d_in[1];   // [1,512,512]
  const float* src   = (const float*)d_in[2];   // [64,64]
  const float* mlr   = (const float*)d_in[3];   // [64,64]
  float* out = (float*)d_out;

  float* w = (float*)d_ws;
  float* CM   = w + 0 * NPIX;
  float* SM   = w + 1 * NPIX;
  float* IBAR = w + 2 * NPIX;
  float* T0   = w + 3 * NPIX;   // P  -> Hr
  float* T1   = w + 4 * NPIX;   // Q  -> Hi
  float* T2   = w + 5 * NPIX;   // Fr
  float* T3   = w + 6 * NPIX;   // Fi
  float* EDGE = w + 7 * NPIX;
  float* AV   = w + 8 * NPIX;
  float* AH   = w + 9 * NPIX;
  float* CV   = w + 10 * NPIX;
  float* CH   = w + 11 * NPIX;
  float* IMGA = w + 12 * NPIX;
  float* IMGB = w + 13 * NPIX;
  float* RATIO = w + 14 * NPIX;
  float* SRCS  = RATIO + 4096;
  float* KEEP  = SRCS + 4096;
  float* SHIFT = KEEP + 4096;

  const dim3 gN(1024), bN(256);
  const dim3 gG(32, 8), bG(128);

  kmin_shift<<<1, 256, 0, stream>>>(src, SHIFT);
  kprep<<<gN, bN, 0, stream>>>(guide, ybic, SHIFT, IMGA, IBAR);
  ksrc<<<16, 256, 0, stream>>>(src, mlr, SHIFT, SRCS, KEEP);
  kdiffs<<<gN, bN, 0, stream>>>(guide, IMGA, AV, AH);
  ktwid<<<gN, bN, 0, stream>>>(CM, SM);

  // Forward DFT: G = IBAR * (C - iS) -> P = IBAR*C, Q = IBAR*S
  gemm2_wmma<0><<<gG, bG, 0, stream>>>(IBAR, CM, 1.0f, nullptr, nullptr, 0.0f, T0, 1.0f);
  gemm2_wmma<0><<<gG, bG, 0, stream>>>(IBAR, SM, 1.0f, nullptr, nullptr, 0.0f, T1, 1.0f);
  // F = (C - iS) * G : Fr = C*P - S*Q ; Fi = -C*Q - S*P
  gemm2_wmma<1><<<gG, bG, 0, stream>>>(CM, T0, 1.0f, SM, T1, -1.0f, T2, 1.0f);
  gemm2_wmma<1><<<gG, bG, 0, stream>>>(CM, T1, -1.0f, SM, T0, -1.0f, T3, 1.0f);
  // high-frequency mask on full grid
  kmask<<<gN, bN, 0, stream>>>(T2, T3);
  // Inverse: H = (C + iS) * F' : Hr = C*Fr' - S*Fi' ; Hi = C*Fi' + S*Fr'
  gemm2_wmma<1><<<gG, bG, 0, stream>>>(CM, T2, 1.0f, SM, T3, -1.0f, T0, 1.0f);
  gemm2_wmma<1><<<gG, bG, 0, stream>>>(CM, T3, 1.0f, SM, T2, 1.0f, T1, 1.0f);
  // edge = Re(H * (C + iS)) / N^2 = (Hr*C - Hi*S) / N^2
  gemm2_wmma<1><<<gG, bG, 0, stream>>>(T0, CM, 1.0f, T1, SM, -1.0f, EDGE,
                                       1.0f / (float)NPIX);

  kcoef<<<gN, bN, 0, stream>>>(AV, AH, EDGE, CV, CH);

  for (int it = 0; it < 64; ++it) {
    kstencil<<<gN, bN, 0, stream>>>(IMGA, CV, CH, IMGB);
    kpool<<<dim3(64, 64), 64, 0, stream>>>(IMGB, SRCS, KEEP, RATIO);
    kscale<<<gN, bN, 0, stream>>>(IMGB, RATIO, IMGA);
  }

  kfinal<<<gN, bN, 0, stream>>>(IMGA, SHIFT, out);
}